// MambaLayer_26130581028930
// MI455X (gfx1250) — compile-verified
//
#include <hip/hip_runtime.h>
#include <hip/hip_bf16.h>

// Mamba layer for MI455X (gfx1250), fp32 end-to-end using V_WMMA_F32_16X16X4_F32.
// Dims: D_MODEL=512, D_INNER=1024, D_STATE=16, D_CONV=4, DT_RANK=32, B=2, L=2048.

#define BATCH   2
#define SEQ     2048
#define DMODEL  512
#define DINNER  1024
#define DSTATE  16
#define DTRANK  32
#define MROWS   (BATCH * SEQ)      // 4096 token rows

typedef float v2f __attribute__((ext_vector_type(2)));
typedef float v8f __attribute__((ext_vector_type(8)));

// ---------------------------------------------------------------------------
// RMSNorm + mask: one block (256 threads) per token row of 512 floats.
// ---------------------------------------------------------------------------
__global__ void rmsnorm_kernel(const float* __restrict__ x,
                               const float* __restrict__ w,
                               const int* __restrict__ mask,
                               float* __restrict__ out) {
    const int row = blockIdx.x;            // 0..4095
    const int tid = threadIdx.x;           // 0..255, 2 elements each
    const float2 v = *(const float2*)(x + (size_t)row * DMODEL + tid * 2);
    float ss = v.x * v.x + v.y * v.y;
    #pragma unroll
    for (int off = 16; off > 0; off >>= 1)
        ss += __shfl_down(ss, off, 32);    // wave32 reduction
    __shared__ float red[8];
    __shared__ float scale_s;
    if ((tid & 31) == 0) red[tid >> 5] = ss;
    __syncthreads();
    if (tid == 0) {
        float tot = 0.f;
        #pragma unroll
        for (int i = 0; i < 8; ++i) tot += red[i];
        scale_s = rsqrtf(tot * (1.0f / (float)DMODEL) + 1e-5f) * (float)mask[row];
    }
    __syncthreads();
    const float sc = scale_s;
    const float2 wv = *(const float2*)(w + tid * 2);
    float2 o;
    o.x = v.x * sc * wv.x;
    o.y = v.y * sc * wv.y;
    *(float2*)(out + (size_t)row * DMODEL + tid * 2) = o;
}

// ---------------------------------------------------------------------------
// fp32 WMMA GEMM:  C[M,N] = A[M,K] (row-major, leading dim lda) * W[N,K]^T
// Wave computes a 32x64 strip: 2 A-fragments x 4 B-fragments -> 8 WMMAs per
// K-step from 6 loads. Fragments are software-pipelined (register double
// buffering) so the next iteration's loads are in flight while the current
// 8 WMMAs execute — no per-WMMA s_wait_loadcnt 0 stalls.
// 4 waves/block -> 128x64 tile per workgroup.
// EPI: 0 = plain, 1 = softplus(val + bias[col]), 2 = resid + mask[row]*val.
// ---------------------------------------------------------------------------
template <int EPI>
__global__ void gemm_wmma_f32(const float* __restrict__ A,
                              const float* __restrict__ W,
                              float* __restrict__ C,
                              int M, int N, int K, int lda,
                              const float* __restrict__ bias,
                              const float* __restrict__ resid,
                              const int* __restrict__ mask) {
    const int lane = threadIdx.x & 31;
    const int wave = threadIdx.x >> 5;     // 0..3
    const int half = lane >> 4;            // 0: K={0,1}, 1: K={2,3} sub-fragment
    const int mn   = lane & 15;            // row/col within fragment
    const int m0   = blockIdx.x * 128 + wave * 32;
    const int n0   = blockIdx.y * 64;

    v8f acc[8] = {};

    const float* ar0 = A + (size_t)(m0 + mn) * lda + 2 * half;        // rows m0..m0+15
    const float* ar1 = A + (size_t)(m0 + 16 + mn) * lda + 2 * half;   // rows m0+16..m0+31
    const float* w0 = W + (size_t)(n0 +  0 + mn) * K + 2 * half;
    const float* w1 = W + (size_t)(n0 + 16 + mn) * K + 2 * half;
    const float* w2 = W + (size_t)(n0 + 32 + mn) * K + 2 * half;
    const float* w3 = W + (size_t)(n0 + 48 + mn) * K + 2 * half;

    // prologue: fragments for k = 0
    v2f a0c = *(const v2f*)(ar0);
    v2f a1c = *(const v2f*)(ar1);
    v2f b0c = *(const v2f*)(w0);
    v2f b1c = *(const v2f*)(w1);
    v2f b2c = *(const v2f*)(w2);
    v2f b3c = *(const v2f*)(w3);

    for (int k = 4; k < K; k += 4) {
        // issue next iteration's loads first — overlap with the 8 WMMAs below
        const v2f a0n = *(const v2f*)(ar0 + k);
        const v2f a1n = *(const v2f*)(ar1 + k);
        const v2f b0n = *(const v2f*)(w0 + k);
        const v2f b1n = *(const v2f*)(w1 + k);
        const v2f b2n = *(const v2f*)(w2 + k);
        const v2f b3n = *(const v2f*)(w3 + k);

        acc[0] = __builtin_amdgcn_wmma_f32_16x16x4_f32(false, a0c, false, b0c,
                                                       (short)0, acc[0], false, false);
        acc[1] = __builtin_amdgcn_wmma_f32_16x16x4_f32(false, a0c, false, b1c,
                                                       (short)0, acc[1], false, false);
        acc[2] = __builtin_amdgcn_wmma_f32_16x16x4_f32(false, a0c, false, b2c,
                                                       (short)0, acc[2], false, false);
        acc[3] = __builtin_amdgcn_wmma_f32_16x16x4_f32(false, a0c, false, b3c,
                                                       (short)0, acc[3], false, false);
        acc[4] = __builtin_amdgcn_wmma_f32_16x16x4_f32(false, a1c, false, b0c,
                                                       (short)0, acc[4], false, false);
        acc[5] = __builtin_amdgcn_wmma_f32_16x16x4_f32(false, a1c, false, b1c,
                                                       (short)0, acc[5], false, false);
        acc[6] = __builtin_amdgcn_wmma_f32_16x16x4_f32(false, a1c, false, b2c,
                                                       (short)0, acc[6], false, false);
        acc[7] = __builtin_amdgcn_wmma_f32_16x16x4_f32(false, a1c, false, b3c,
                                                       (short)0, acc[7], false, false);

        a0c = a0n; a1c = a1n;
        b0c = b0n; b1c = b1n; b2c = b2n; b3c = b3n;
    }

    // epilogue K-step
    acc[0] = __builtin_amdgcn_wmma_f32_16x16x4_f32(false, a0c, false, b0c,
                                                   (short)0, acc[0], false, false);
    acc[1] = __builtin_amdgcn_wmma_f32_16x16x4_f32(false, a0c, false, b1c,
                                                   (short)0, acc[1], false, false);
    acc[2] = __builtin_amdgcn_wmma_f32_16x16x4_f32(false, a0c, false, b2c,
                                                   (short)0, acc[2], false, false);
    acc[3] = __builtin_amdgcn_wmma_f32_16x16x4_f32(false, a0c, false, b3c,
                                                   (short)0, acc[3], false, false);
    acc[4] = __builtin_amdgcn_wmma_f32_16x16x4_f32(false, a1c, false, b0c,
                                                   (short)0, acc[4], false, false);
    acc[5] = __builtin_amdgcn_wmma_f32_16x16x4_f32(false, a1c, false, b1c,
                                                   (short)0, acc[5], false, false);
    acc[6] = __builtin_amdgcn_wmma_f32_16x16x4_f32(false, a1c, false, b2c,
                                                   (short)0, acc[6], false, false);
    acc[7] = __builtin_amdgcn_wmma_f32_16x16x4_f32(false, a1c, false, b3c,
                                                   (short)0, acc[7], false, false);

    // C/D layout: VGPR r, lanes 0-15 -> M=r, lanes 16-31 -> M=r+8; N = lane%16.
    #pragma unroll
    for (int g = 0; g < 2; ++g) {          // A-fragment group (rows +0 / +16)
        #pragma unroll
        for (int t = 0; t < 4; ++t) {      // N sub-tile
            #pragma unroll
            for (int r = 0; r < 8; ++r) {
                const int row = m0 + g * 16 + r + 8 * half;
                const int col = n0 + t * 16 + mn;
                float v = acc[g * 4 + t][r];
                if (EPI == 1) {
                    v += bias[col];
                    v = (v > 20.f) ? v : log1pf(__expf(v));   // softplus
                }
                if (EPI == 2) {
                    v = resid[(size_t)row * N + col] + (float)mask[row] * v;
                }
                C[(size_t)row * N + col] = v;
            }
        }
    }
}

// ---------------------------------------------------------------------------
// Causal depthwise conv (width 4) + SiLU. u lives in cols [0,1024) of xz.
// ---------------------------------------------------------------------------
__global__ void conv_silu_kernel(const float* __restrict__ xz,
                                 const float* __restrict__ cw,
                                 const float* __restrict__ cb,
                                 float* __restrict__ uc) {
    const int idx = blockIdx.x * blockDim.x + threadIdx.x;  // 4096*1024
    const int d   = idx & (DINNER - 1);
    const int row = idx >> 10;              // b*L + l
    const int l   = row & (SEQ - 1);
    const size_t base = (size_t)row * (2 * DINNER) + d;
    float acc = cb[d];
    #pragma unroll
    for (int j = 0; j < 4; ++j) {
        const int ll = l - 3 + j;
        const float uv = (ll >= 0) ? xz[base + (size_t)(j - 3) * (2 * DINNER)] : 0.f;
        acc += cw[d * 4 + j] * uv;
    }
    uc[idx] = acc / (1.f + __expf(-acc));   // SiLU
}

// ---------------------------------------------------------------------------
// Selective scan: one thread per (b,d) channel, 16-state recurrence in regs,
// strictly sequential over L. B/C per step are block-uniform b128 loads.
// ---------------------------------------------------------------------------
__global__ void scan_kernel(const float* __restrict__ dtb,
                            const float* __restrict__ uc,
                            const float* __restrict__ xdbc,
                            const float* __restrict__ xz,
                            const float* __restrict__ A_log,
                            const float* __restrict__ Dp,
                            float* __restrict__ yg) {
    const int tid = blockIdx.x * blockDim.x + threadIdx.x;  // 0..2047
    const int b = tid >> 10;
    const int d = tid & (DINNER - 1);

    float Aa[DSTATE];
    #pragma unroll
    for (int s = 0; s < DSTATE; ++s) Aa[s] = -__expf(A_log[d * DSTATE + s]);
    const float Dd = Dp[d];

    float h[DSTATE];
    #pragma unroll
    for (int s = 0; s < DSTATE; ++s) h[s] = 0.f;

    const size_t rowBase = (size_t)b * SEQ;
    for (int l = 0; l < SEQ; ++l) {
        const size_t r = rowBase + l;
        const float dt = dtb[r * DINNER + d];
        const float u  = uc[r * DINNER + d];
        const float4* bc = (const float4*)(xdbc + r * 64 + DTRANK);
        const float4 B0 = bc[0], B1 = bc[1], B2 = bc[2], B3 = bc[3];
        const float4 C0 = bc[4], C1 = bc[5], C2 = bc[6], C3 = bc[7];
        const float Bv[DSTATE] = {B0.x, B0.y, B0.z, B0.w, B1.x, B1.y, B1.z, B1.w,
                                  B2.x, B2.y, B2.z, B2.w, B3.x, B3.y, B3.z, B3.w};
        const float Cv[DSTATE] = {C0.x, C0.y, C0.z, C0.w, C1.x, C1.y, C1.z, C1.w,
                                  C2.x, C2.y, C2.z, C2.w, C3.x, C3.y, C3.z, C3.w};
        const float dtu = dt * u;
        float y = 0.f;
        #pragma unroll
        for (int s = 0; s < DSTATE; ++s) {
            const float dA = __expf(dt * Aa[s]);
            h[s] = dA * h[s] + dtu * Bv[s];
            y += h[s] * Cv[s];
        }
        const float yv = y + Dd * u;
        const float z = xz[r * (2 * DINNER) + DINNER + d];
        const float sig = 1.f / (1.f + __expf(-z));
        yg[r * DINNER + d] = yv * (z * sig);     // gate with silu(z)
    }
}

// ---------------------------------------------------------------------------
extern "C" void kernel_launch(void* const* d_in, const int* in_sizes, int n_in,
                              void* d_out, int out_size, void* d_ws, size_t ws_size,
                              hipStream_t stream) {
    const float* x          = (const float*)d_in[0];
    const int*   amask      = (const int*)d_in[1];
    const float* norm_w     = (const float*)d_in[2];
    const float* in_proj_w  = (const float*)d_in[3];   // (2048, 512)
    const float* conv_w     = (const float*)d_in[4];   // (1024, 1, 4)
    const float* conv_b     = (const float*)d_in[5];
    const float* x_proj_w   = (const float*)d_in[6];   // (64, 1024)
    const float* dt_proj_w  = (const float*)d_in[7];   // (1024, 32)
    const float* dt_proj_b  = (const float*)d_in[8];
    const float* A_log      = (const float*)d_in[9];
    const float* Dp         = (const float*)d_in[10];
    const float* out_proj_w = (const float*)d_in[11];  // (512, 1024)
    float* out = (float*)d_out;

    // Workspace carve-up (all fp32), ~90 MB total.
    float* ws   = (float*)d_ws;
    float* hn   = ws;                                   // 4096 x 512
    float* xz   = hn   + (size_t)MROWS * DMODEL;        // 4096 x 2048 (u | z)
    float* uc   = xz   + (size_t)MROWS * 2 * DINNER;    // 4096 x 1024 conv+silu
    float* xdbc = uc   + (size_t)MROWS * DINNER;        // 4096 x 64  (dt_r|B|C)
    float* dtb  = xdbc + (size_t)MROWS * 64;            // 4096 x 1024 softplus dt
    float* yg   = dtb  + (size_t)MROWS * DINNER;        // 4096 x 1024 gated y

    // 1. RMSNorm + mask
    rmsnorm_kernel<<<MROWS, 256, 0, stream>>>(x, norm_w, amask, hn);

    // 2. in_proj: (4096x512) x (2048x512)^T -> xz
    gemm_wmma_f32<0><<<dim3(MROWS / 128, 2 * DINNER / 64), 128, 0, stream>>>(
        hn, in_proj_w, xz, MROWS, 2 * DINNER, DMODEL, DMODEL,
        nullptr, nullptr, nullptr);

    // 3. causal depthwise conv + SiLU on u
    conv_silu_kernel<<<(MROWS * DINNER) / 256, 256, 0, stream>>>(xz, conv_w, conv_b, uc);

    // 4. x_proj: (4096x1024) x (64x1024)^T -> xdbc
    gemm_wmma_f32<0><<<dim3(MROWS / 128, 64 / 64), 128, 0, stream>>>(
        uc, x_proj_w, xdbc, MROWS, 64, DINNER, DINNER,
        nullptr, nullptr, nullptr);

    // 5. dt = softplus(dt_r x dt_proj_w^T + b); A = xdbc with lda=64, K=32
    gemm_wmma_f32<1><<<dim3(MROWS / 128, DINNER / 64), 128, 0, stream>>>(
        xdbc, dt_proj_w, dtb, MROWS, DINNER, DTRANK, 64,
        dt_proj_b, nullptr, nullptr);

    // 6. selective scan + skip + gating (2048 channel threads, 8 blocks)
    scan_kernel<<<(BATCH * DINNER) / 256, 256, 0, stream>>>(
        dtb, uc, xdbc, xz, A_log, Dp, yg);

    // 7. out_proj + mask + residual: out = x + mask * (yg x out_proj_w^T)
    gemm_wmma_f32<2><<<dim3(MROWS / 128, DMODEL / 64), 128, 0, stream>>>(
        yg, out_proj_w, out, MROWS, DMODEL, DINNER, DINNER,
        nullptr, x, amask);
}